// projection_68161130987644
// MI455X (gfx1250) — compile-verified
//
#include <hip/hip_runtime.h>
#include <cstdint>
#include <cstddef>

// Cone-beam DRR projector for MI455X (gfx1250), double-buffered async pipeline.
//
// Geometry facts:
//   * sample k lies exactly on volume plane y=k  -> trilinear ==
//     0.5*(bilinear(slab k-1) + bilinear(slab k)) with identical (iz,ix) coords
//   * per-pixel slab coordinates are linear in k: iz = az*k + bz, ix = ax*k + bx
//
// Data path: volume (8MB) is L2-resident. Per (16x16 detector tile, k) a
// 21x24 window of slabs k-1 and k is staged into LDS with exactly one
// global_load_async_to_lds_b128 per thread per stage (wave-uniform ASYNCcnt),
// double-buffered so stage k+1 copies overlap stage k compute; s_wait_asynccnt 1
// retires stage k (async loads complete in order) while k+1 stays in flight.

#define DVOL   128
#define NPROJ  10
#define RESD   179
#define RESH   179
#define TILE   16
#define TPB    256
#define WINZ   21            // window rows (z)
#define WINX   24            // window cols (x), multiple of 4 floats (16B)
#define SLABF  (WINZ * WINX) // 504 floats per staged slab
#define TRPS   126           // b128 transfers per slab = (WINX/4)*WINZ

__device__ __forceinline__ int iclamp(int v, int lo, int hi) {
    return v < lo ? lo : (v > hi ? hi : v);
}

// One b128 async copy per thread: threads [0,126) load slab y=k, [126,256)
// load slab y=k-1 (clamped -> a few duplicate transfers, harmless, keeps the
// per-wave ASYNCcnt increment exactly 1 per stage for precise pipelining).
__device__ __forceinline__ void issue_stage(const float* __restrict__ vol, int k,
                                            int wz0, int wx0, uint32_t buf_base,
                                            int t) {
    int sel = (t < TRPS) ? 1 : 0;                 // 1 = slab y=k, 0 = slab y=k-1
    int e   = (t < TRPS) ? t : (t - TRPS);
    e = iclamp(e, 0, TRPS - 1);
    if (k == 0) sel = 1;                          // no slab -1; duplicates of slab 0
    const int y  = k - 1 + sel;
    const int r  = e / 6;                         // z-row within window
    const int c4 = (e - r * 6) * 4;               // x-col (float) within window
    const uint32_t loff = buf_base + (uint32_t)(((sel ? SLABF : 0) + r * WINX + c4) * 4);
    const float* gp = vol + ((size_t)(wz0 + r) * (DVOL * DVOL)
                             + (size_t)y * DVOL + (size_t)(wx0 + c4));
    asm volatile("global_load_async_to_lds_b128 %0, %1, off"
                 :: "v"(loff), "v"((uint64_t)(uintptr_t)gp)
                 : "memory");
}

__global__ __launch_bounds__(TPB)
void drr_project_kernel(const float* __restrict__ vol,     // [128,128,128] (z=D, y=W, x=H)
                        const float* __restrict__ poses,   // [50,3]
                        const int*   __restrict__ idx,     // [10]
                        float*       __restrict__ out) {   // [10,179,179]
    __shared__ float lds[4 * SLABF];  // 2 stages x {slab k-1, slab k}

    const int p     = blockIdx.z;
    const int tileX = blockIdx.x;      // detector gx tile (RES_D axis)
    const int tileY = blockIdx.y;      // detector gy tile (RES_H axis)
    const int t     = threadIdx.x;
    const int gx    = tileX * TILE + (t & (TILE - 1));
    const int gy    = tileY * TILE + (t >> 4);

    const int   e  = idx[p];
    const float ex = poses[e * 3 + 0];
    const float ey = poses[e * 3 + 1];   // in [256,384): well away from the volume
    const float ez = poses[e * 3 + 2];

    // detector pixel on y=0 plane; linspace(-89.5, 88.5, 179), step 1
    const float gxv = -89.5f + (float)gx;
    const float gyv = -89.5f + (float)gy;
    const float vx = gxv - ex, vy = -ey, vz = gyv - ez;
    const float inv_vy = 1.0f / vy;                 // uniform (vy = -ey for all pixels)
    const float az = vx * inv_vy;                   // d(iz)/dk   (volume D axis)
    const float ax = vz * inv_vy;                   // d(ix)/dk   (volume H axis)
    const float bz = fmaf(-az, ey, ex + 63.5f);     // iz(k) = az*k + bz
    const float bx = fmaf(-ax, ey, ez + 63.5f);
    const float dxv = sqrtf(vx * vx + vy * vy + vz * vz) * fabsf(inv_vy);

    // workgroup-uniform corner ray: iz is monotone increasing in gxv
    // (slope (ey-k)/ey > 0), so the tile corner gives the per-step minimum.
    const float caz = ((-89.5f + (float)(tileX * TILE)) - ex) * inv_vy;
    const float cax = ((-89.5f + (float)(tileY * TILE)) - ez) * inv_vy;
    const float cbz = fmaf(-caz, ey, ex + 63.5f);
    const float cbx = fmaf(-cax, ey, ez + 63.5f);

    const uint32_t lds_base = (uint32_t)(uintptr_t)&lds[0]; // low 32b of flat = LDS offset

    // window origin for stage k (uniform per workgroup); -1 guard row absorbs
    // FP rounding; x origin aligned down to 4 floats for b128 transfers.
    auto origin_z = [&](int k) {
        return iclamp((int)floorf(fmaf(caz, (float)k, cbz)) - 1, 0, DVOL - WINZ);
    };
    auto origin_x = [&](int k) {
        return iclamp((int)floorf(fmaf(cax, (float)k, cbx)) - 1, 0, DVOL - WINX) & ~3;
    };

    // ---- prologue: issue stage 0 into buffer 0 ----
    int wz0c = origin_z(0);
    int wx0c = origin_x(0);
    issue_stage(vol, 0, wz0c, wx0c, lds_base, t);

    float acc = 0.0f;
    for (int k = 0; k < DVOL; ++k) {
        int wz0n = 0, wx0n = 0;
        if (k < DVOL - 1) {
            // issue stage k+1 into the other buffer, then retire stage k only
            wz0n = origin_z(k + 1);
            wx0n = origin_x(k + 1);
            issue_stage(vol, k + 1, wz0n, wx0n,
                        lds_base + (uint32_t)(((k + 1) & 1) * 2 * SLABF * 4), t);
            asm volatile("s_wait_asynccnt 0x1" ::: "memory");
        } else {
            asm volatile("s_wait_asynccnt 0x0" ::: "memory");
        }
        __syncthreads();   // stage-k windows visible to all waves

        // ---- bilinear taps on slabs k-1 and k at (iz, ix) ----
        const float kf  = (float)k;
        const float izf = fmaf(az, kf, bz);
        const float ixf = fmaf(ax, kf, bx);
        const float z0f = floorf(izf), x0f = floorf(ixf);
        const float tz  = izf - z0f,  tx  = ixf - x0f;
        const int z0 = (int)z0f, x0 = (int)x0f;
        const int z1 = z0 + 1,   x1 = x0 + 1;

        const bool vzl = (unsigned)z0 < (unsigned)DVOL;
        const bool vzh = (unsigned)z1 < (unsigned)DVOL;
        const bool vxl = (unsigned)x0 < (unsigned)DVOL;
        const bool vxh = (unsigned)x1 < (unsigned)DVOL;

        // local window coords; clamp only bites on invalid (masked) corners
        const int lz0 = iclamp(z0 - wz0c, 0, WINZ - 1);
        const int lz1 = iclamp(z1 - wz0c, 0, WINZ - 1);
        const int lx0 = iclamp(x0 - wx0c, 0, WINX - 1);
        const int lx1 = iclamp(x1 - wx0c, 0, WINX - 1);

        const float* sb = &lds[((k & 1) * 2 + 1) * SLABF];  // slab y=k
        float v00 = (vzl && vxl) ? sb[lz0 * WINX + lx0] : 0.0f;
        float v01 = (vzl && vxh) ? sb[lz0 * WINX + lx1] : 0.0f;
        float v10 = (vzh && vxl) ? sb[lz1 * WINX + lx0] : 0.0f;
        float v11 = (vzh && vxh) ? sb[lz1 * WINX + lx1] : 0.0f;
        if (k > 0) {                                        // slab y=k-1 (zero-pad at k=0)
            const float* sa = &lds[((k & 1) * 2) * SLABF];
            v00 += (vzl && vxl) ? sa[lz0 * WINX + lx0] : 0.0f;
            v01 += (vzl && vxh) ? sa[lz0 * WINX + lx1] : 0.0f;
            v10 += (vzh && vxl) ? sa[lz1 * WINX + lx0] : 0.0f;
            v11 += (vzh && vxh) ? sa[lz1 * WINX + lx1] : 0.0f;
        }
        const float w00 = (1.0f - tz) * (1.0f - tx);
        const float w01 = (1.0f - tz) * tx;
        const float w10 = tz * (1.0f - tx);
        const float w11 = tz * tx;
        const float s = fmaf(w00, v00, fmaf(w01, v01, fmaf(w10, v10, w11 * v11)));

        acc = fmaf(0.5f, s, acc);   // ty == 0.5 exactly (sample plane y == k)

        __syncthreads();   // readers done before next iteration overwrites a buffer
        wz0c = wz0n;
        wx0c = wx0n;
    }

    if (gx < RESD && gy < RESH) {
        out[((size_t)p * RESD + gx) * RESH + gy] = acc * dxv;
    }
}

// second tuple output: pass idx through (bit-exact int32 into the tail)
__global__ void copy_idx_kernel(const int* __restrict__ idx, int* __restrict__ out_tail) {
    const int i = threadIdx.x;
    if (i < NPROJ) out_tail[i] = idx[i];
}

extern "C" void kernel_launch(void* const* d_in, const int* in_sizes, int n_in,
                              void* d_out, int out_size, void* d_ws, size_t ws_size,
                              hipStream_t stream) {
    (void)in_sizes; (void)n_in; (void)d_ws; (void)ws_size; (void)out_size;
    const float* vol   = (const float*)d_in[0];   // I_rec [1,1,128,128,128]
    const float* poses = (const float*)d_in[1];   // [50,3]
    const int*   idx   = (const int*)d_in[2];     // [10]
    float*       out   = (float*)d_out;

    dim3 grid((RESD + TILE - 1) / TILE,   // 12
              (RESH + TILE - 1) / TILE,   // 12
              NPROJ);                     // 10
    drr_project_kernel<<<grid, TPB, 0, stream>>>(vol, poses, idx, out);

    int* tail = (int*)(out + (size_t)NPROJ * RESD * RESH);
    copy_idx_kernel<<<1, 32, 0, stream>>>(idx, tail);
}